// MonodomainSolverPINNs_61641370632809
// MI455X (gfx1250) — compile-verified
//
#include <hip/hip_runtime.h>

// ---------------------------------------------------------------------------
// Fused forward-mode PINN residual for the monodomain MLP (gfx1250).
// 6 Taylor channels (h, g0, g1, gt, s0, s1) propagated through 4 hidden
// layers with f32 WMMA (v_wmma_f32_16x16x4_f32). One 16-point tile per
// workgroup; activations resident in LDS for the whole network; weight
// K-panels double-buffered in LDS and staged with GLOBAL_LOAD_ASYNC_TO_LDS
// (ASYNCcnt) so L2 latency overlaps the WMMA stream.
// ---------------------------------------------------------------------------

typedef __attribute__((ext_vector_type(2))) float v2f;
typedef __attribute__((ext_vector_type(8))) float v8f;
typedef int v4i_t __attribute__((__vector_size__(4 * sizeof(int))));

#define HID     256
#define NLAYERS 4
#define NPTS    131072
#define MTILE   16
#define KP      64          // K-panel of weights staged in LDS
#define NPANELS (HID / KP)  // 4
#define APITCH  260         // padded activation pitch (bank-conflict-free A frags)
#define WPITCH  260         // padded weight-panel pitch

#define ACT_FLOATS (6 * MTILE * APITCH)   // 24960 floats
#define WP_FLOATS  (KP * WPITCH)          // 16640 floats
#define SMEM_FLOATS (ACT_FLOATS + 2 * WP_FLOATS)  // ~233 KB of 320 KB WGP LDS

#if __has_builtin(__builtin_amdgcn_global_load_async_to_lds_b128) && \
    __has_builtin(__builtin_amdgcn_s_wait_asynccnt)
#define ASYNC_STAGE 1
typedef __attribute__((address_space(1))) v4i_t* gvec_ptr;
typedef __attribute__((address_space(3))) v4i_t* svec_ptr;
#else
#define ASYNC_STAGE 0
#endif

__device__ __forceinline__ v8f wmma_f32(v2f a, v2f b, v8f c) {
  // 8-arg pattern: (neg_a, A, neg_b, B, c_mod, C, reuse_a, reuse_b)
  return __builtin_amdgcn_wmma_f32_16x16x4_f32(
      false, a, false, b, (short)0, c, false, false);
}

// Stage W[k0:k0+KP, 0:256] into an LDS panel. 256 threads, 16 b128 each.
__device__ __forceinline__ void stage_panel(float* __restrict__ wpbuf,
                                            const float* __restrict__ Wsrc,
                                            int k0, int tid) {
  const int slot = tid & 63;   // float4 slot within a 256-wide row
  const int roff = tid >> 6;   // 4 rows per pass
#pragma unroll
  for (int rr = 0; rr < KP; rr += 4) {
    const int kk = rr + roff;
#if ASYNC_STAGE
    __builtin_amdgcn_global_load_async_to_lds_b128(
        (gvec_ptr)&Wsrc[(k0 + kk) * HID + slot * 4],
        (svec_ptr)&wpbuf[kk * WPITCH + slot * 4],
        0, 0);
#else
    const float4 v = *(const float4*)&Wsrc[(k0 + kk) * HID + slot * 4];
    *(float4*)&wpbuf[kk * WPITCH + slot * 4] = v;
#endif
  }
}

__global__ __launch_bounds__(256, 1)
void pinn_fused(const float* __restrict__ X,
                const float* __restrict__ Win,
                const float* __restrict__ bin,
                const float* __restrict__ Wh,
                const float* __restrict__ bh,
                const float* __restrict__ Wout,
                float* __restrict__ partial) {
  extern __shared__ float smem[];
  float* act = smem;                       // [6][MTILE][APITCH]
  float* wpbase = smem + ACT_FLOATS;       // [2][KP][WPITCH]

  const int tid  = threadIdx.x;
  const int lane = tid & 31;
  const int wid  = tid >> 5;           // 0..7
  const int mrow = lane & 15;          // A-fragment row (point)
  const int ncol = lane & 15;          // B/C column within 16-wide tile
  const int kb   = (lane >> 4) << 1;   // K sub-offset {0,2} per half-wave
  const int p0   = blockIdx.x * MTILE;

  // Kick off layer-0 panel-0 weight staging immediately (overlaps input layer)
  stage_panel(wpbase, Wh, 0, tid);

  // -------------------- input layer (3 -> 256), forward-mode seed ----------
  {
    const int m  = tid >> 4;   // point 0..15
    const int ng = tid & 15;   // 16-column group
    const float x0 = X[(p0 + m) * 3 + 0];
    const float x1 = X[(p0 + m) * 3 + 1];
    const float x2 = X[(p0 + m) * 3 + 2];
#pragma unroll
    for (int j = 0; j < 16; ++j) {
      const int n = ng * 16 + j;
      const float w0 = Win[0 * HID + n];
      const float w1 = Win[1 * HID + n];
      const float w2 = Win[2 * HID + n];
      const float z  = x0 * w0 + x1 * w1 + x2 * w2 + bin[n];
      const float h  = tanhf(z);
      const float u  = 1.f - h * h;       // tanh'
      const float wc = -2.f * h * u;      // tanh''
      act[(0 * MTILE + m) * APITCH + n] = h;            // value
      act[(1 * MTILE + m) * APITCH + n] = u * w0;       // d/dx0
      act[(2 * MTILE + m) * APITCH + n] = u * w1;       // d/dx1
      act[(3 * MTILE + m) * APITCH + n] = u * w2;       // d/dt
      act[(4 * MTILE + m) * APITCH + n] = wc * w0 * w0; // d2/dx0^2
      act[(5 * MTILE + m) * APITCH + n] = wc * w1 * w1; // d2/dx1^2
    }
  }
  __syncthreads();

  const v8f zero8 = {0.f, 0.f, 0.f, 0.f, 0.f, 0.f, 0.f, 0.f};

  // -------------------- 4 hidden layers, fused ----------------------------
  for (int l = 0; l < NLAYERS; ++l) {
    const float* __restrict__ Wl = Wh + l * HID * HID;
    const float* __restrict__ bl = bh + l * HID;

    v8f acc[6][2];
#pragma unroll
    for (int c = 0; c < 6; ++c)
#pragma unroll
      for (int t2 = 0; t2 < 2; ++t2) acc[c][t2] = zero8;

    for (int p = 0; p < NPANELS; ++p) {
      const int k0  = p * KP;
      const int cur = p & 1;
      float* wp  = wpbase + cur * WP_FLOATS;
      float* nxt = wpbase + (1 - cur) * WP_FLOATS;

      // Issue next panel (or next layer's panel 0) into the other buffer.
      // That buffer was last *read* during iteration p-1, and the barrier at
      // the end of p-1's compute guarantees every wave is done with it.
      if (p < NPANELS - 1) {
        stage_panel(nxt, Wl, k0 + KP, tid);
      } else if (l < NLAYERS - 1) {
        stage_panel(nxt, Wl + HID * HID, 0, tid);
      }
#if ASYNC_STAGE
      // Next panel (if any) keeps 16 async loads in flight; retire current.
      if (p == NPANELS - 1 && l == NLAYERS - 1) {
        __builtin_amdgcn_s_wait_asynccnt(0);
      } else {
        __builtin_amdgcn_s_wait_asynccnt(16);
      }
#endif
      __syncthreads();   // current panel visible to all waves

#pragma unroll 4
      for (int k = 0; k < KP; k += 4) {
        v2f afrag[6];
#pragma unroll
        for (int c = 0; c < 6; ++c)
          afrag[c] = *(const v2f*)&act[(c * MTILE + mrow) * APITCH + k + kb];
        v2f bfrag[2];
#pragma unroll
        for (int t2 = 0; t2 < 2; ++t2) {
          const int cb = wid * 32 + t2 * 16 + ncol;
          v2f b;
          b.x = wp[(k + kb + 0) * WPITCH + cb];
          b.y = wp[(k + kb + 1) * WPITCH + cb];
          bfrag[t2] = b;
        }
#pragma unroll
        for (int c = 0; c < 6; ++c)
#pragma unroll
          for (int t2 = 0; t2 < 2; ++t2)
            acc[c][t2] = wmma_f32(afrag[c], bfrag[t2], acc[c][t2]);
      }
      __syncthreads();   // compute done before this buffer is restaged
    }

    // ---- elementwise tanh chain in registers; write next-layer channels ---
#pragma unroll
    for (int t2 = 0; t2 < 2; ++t2) {
      const int ct   = wid * 32 + t2 * 16;   // column base of this tile
      const float bc = bl[ct + ncol];
      const int mhi  = (lane >> 4) * 8;      // C layout: VGPR j -> M = mhi+j
#pragma unroll
      for (int j = 0; j < 8; ++j) {
        const int m = mhi + j;
        const float z   = acc[0][t2][j] + bc;
        const float zg0 = acc[1][t2][j];
        const float zg1 = acc[2][t2][j];
        const float zgt = acc[3][t2][j];
        const float zs0 = acc[4][t2][j];
        const float zs1 = acc[5][t2][j];
        const float a  = tanhf(z);
        const float u  = 1.f - a * a;
        const float wc = -2.f * a * u;
        act[(0 * MTILE + m) * APITCH + ct + ncol] = a;
        act[(1 * MTILE + m) * APITCH + ct + ncol] = u * zg0;
        act[(2 * MTILE + m) * APITCH + ct + ncol] = u * zg1;
        act[(3 * MTILE + m) * APITCH + ct + ncol] = u * zgt;
        act[(4 * MTILE + m) * APITCH + ct + ncol] = u * zs0 + wc * zg0 * zg0;
        act[(5 * MTILE + m) * APITCH + ct + ncol] = u * zs1 + wc * zg1 * zg1;
      }
    }
    __syncthreads();
  }

  // ---- output layer: residual = (gt - s0 - s1) . W_out  (M_DIFF = 1) ------
  {
    const int m    = tid >> 4;
    const int part = tid & 15;
    float s = 0.f;
#pragma unroll
    for (int j = 0; j < 16; ++j) {
      const int n = part * 16 + j;
      const float v = act[(3 * MTILE + m) * APITCH + n]
                    - act[(4 * MTILE + m) * APITCH + n]
                    - act[(5 * MTILE + m) * APITCH + n];
      s += v * Wout[n];
    }
#pragma unroll
    for (int off = 8; off > 0; off >>= 1) s += __shfl_xor(s, off, 16);
    if (part == 0) wpbase[m] = s * s;   // scratch reuse (post-barrier)
  }
  __syncthreads();
  if (tid == 0) {
    float tot = 0.f;
#pragma unroll
    for (int i = 0; i < 16; ++i) tot += wpbase[i];
    partial[blockIdx.x] = tot;
  }
}

__global__ __launch_bounds__(256, 1)
void pinn_reduce(const float* __restrict__ partial, float* __restrict__ out,
                 int n) {
  __shared__ float s[256];
  float acc = 0.f;
  for (int i = threadIdx.x; i < n; i += 256) acc += partial[i];
  s[threadIdx.x] = acc;
  __syncthreads();
  for (int off = 128; off > 0; off >>= 1) {
    if (threadIdx.x < off) s[threadIdx.x] += s[threadIdx.x + off];
    __syncthreads();
  }
  if (threadIdx.x == 0) {
    out[0] = s[0] / (float)NPTS;  // pde_loss
    out[1] = 0.f;                 // ode_loss
  }
}

extern "C" void kernel_launch(void* const* d_in, const int* in_sizes, int n_in,
                              void* d_out, int out_size, void* d_ws,
                              size_t ws_size, hipStream_t stream) {
  const float* X    = (const float*)d_in[0];  // [131072,3]
  const float* Win  = (const float*)d_in[1];  // [3,256]
  const float* bin  = (const float*)d_in[2];  // [256]
  const float* Wh   = (const float*)d_in[3];  // [4,256,256]
  const float* bh   = (const float*)d_in[4];  // [4,256]
  const float* Wout = (const float*)d_in[5];  // [256,1]
  // d_in[6] = b_out: constants vanish under differentiation; unused.
  float* partial = (float*)d_ws;              // 8192 floats
  float* out     = (float*)d_out;             // [pde_loss, ode_loss]

  const int ntiles = NPTS / MTILE;            // 8192 workgroups
  const size_t smem = SMEM_FLOATS * sizeof(float);
  pinn_fused<<<ntiles, 256, smem, stream>>>(X, Win, bin, Wh, bh, Wout, partial);
  pinn_reduce<<<1, 256, 0, stream>>>(partial, out, ntiles);
}